// CosineRouter_9620726743475
// MI455X (gfx1250) — compile-verified
//
#include <hip/hip_runtime.h>
#include <hip/hip_bf16.h>

typedef __attribute__((ext_vector_type(2))) float v2f;
typedef __attribute__((ext_vector_type(8))) float v8f;
typedef __attribute__((ext_vector_type(4))) int   v4i;

#define N_ROWS  16384
#define DIM     2048
#define RDIM    16
#define NEXP    64
#define ROWS_PER_BLOCK 128
#define THREADS 256
#define KTILE   32

// LDS layout (floats), regions overlapped by lifetime:
//   phase A (main GEMM, double buffered): xs0 [0,4608) xs1 [4608,9216) ws0 [9216,9792) ws1 [9792,10368)
//   phase C/D (scores):                   sc  [0,8320)   (reuses tile buffers)
//   persistent:                           qs  [10368,12544)   kn [12544,13632)
#define XS_STRIDE 36
#define WS_STRIDE 36
#define XS0_OFF  0
#define XS1_OFF  4608
#define WS0_OFF  9216
#define WS1_OFF  9792
#define SC_OFF   0
#define SC_STRIDE 65
#define QS_OFF   10368
#define QS_STRIDE 17
#define KN_OFF   12544
#define KN_STRIDE 17
#define SMEM_FLOATS 13632

#define GLOBAL_AS __attribute__((address_space(1)))
#define LDS_AS    __attribute__((address_space(3)))

// CDNA5 async memory->LDS copy (GLOBAL_LOAD_ASYNC_TO_LDS_B128, tracked by ASYNCcnt)
__device__ __forceinline__ void async_copy_b128(const float* gsrc, float* lds_dst) {
#if __has_builtin(__builtin_amdgcn_global_load_async_to_lds_b128)
    __builtin_amdgcn_global_load_async_to_lds_b128(
        (GLOBAL_AS v4i*)gsrc, (LDS_AS v4i*)lds_dst, 0, 0);
#else
    unsigned int loff = (unsigned int)(unsigned long long)(LDS_AS void*)lds_dst;
    asm volatile("global_load_async_to_lds_b128 %0, %1, off"
                 :: "v"(loff), "v"(gsrc) : "memory");
#endif
}

__device__ __forceinline__ void wait_async0() {
#if __has_builtin(__builtin_amdgcn_s_wait_asynccnt)
    __builtin_amdgcn_s_wait_asynccnt(0);
#else
    asm volatile("s_wait_asynccnt 0" ::: "memory");
#endif
}

__global__ __launch_bounds__(THREADS)
void cosine_router_kernel(const float* __restrict__ x,
                          const float* __restrict__ Wq,
                          const float* __restrict__ keys,
                          int*   __restrict__ out_idx,
                          float* __restrict__ out_w,
                          float* __restrict__ out_scores) {
    __shared__ float smem[SMEM_FLOATS];
    float* sc = smem + SC_OFF;
    float* qs = smem + QS_OFF;
    float* kn = smem + KN_OFF;

    const int tid  = threadIdx.x;
    const int wid  = tid >> 5;
    const int lane = tid & 31;
    const int ln   = lane & 15;
    const int hi   = lane >> 4;
    const int blockRow = blockIdx.x * ROWS_PER_BLOCK;

    // per-thread staging coordinates (branch-free: 4 x-tile copies each)
    const int srow = tid >> 3;      // 0..31 (x rows, 4 strided groups of 32)
    const int scg  = tid & 7;       // column group: 8 x float4 = 32 floats

    // ---- prefetch tile 0 (async -> LDS, double buffer 0) ----
    {
        float* xb = smem + XS0_OFF;
        float* wb = smem + WS0_OFF;
        #pragma unroll
        for (int u = 0; u < 4; ++u) {
            int row = srow + 32 * u;
            async_copy_b128(x + (size_t)(blockRow + row) * DIM + scg * 4,
                            xb + row * XS_STRIDE + scg * 4);
        }
        if (tid < RDIM * 8) {
            async_copy_b128(Wq + (size_t)srow * DIM + scg * 4,
                            wb + srow * WS_STRIDE + scg * 4);
        }
    }

    // ---- normalize keys into LDS (tiny: 64x16) while prefetch is in flight ----
    if (tid < NEXP) {
        float v[RDIM];
        float s = 0.f;
        #pragma unroll
        for (int r = 0; r < RDIM; ++r) { v[r] = keys[tid * RDIM + r]; s += v[r] * v[r]; }
        float inv = 1.0f / fmaxf(sqrtf(s), 1e-12f);
        #pragma unroll
        for (int r = 0; r < RDIM; ++r) kn[tid * KN_STRIDE + r] = v[r] * inv;
    }

    wait_async0();
    __syncthreads();

    // ---- main GEMM: q_raw = x @ Wq^T via V_WMMA_F32_16X16X4_F32, double buffered ----
    v8f acc = {};
    const int mbase = wid * 16;
    for (int k0 = 0; k0 < DIM; k0 += KTILE) {
        const int b = (k0 >> 5) & 1;
        const float* xb = smem + (b ? XS1_OFF : XS0_OFF);
        const float* wb = smem + (b ? WS1_OFF : WS0_OFF);

        // prefetch next tile into the other buffer (overlaps with WMMA below)
        const int knext = k0 + KTILE;
        if (knext < DIM) {
            float* xn = smem + (b ? XS0_OFF : XS1_OFF);
            float* wn = smem + (b ? WS0_OFF : WS1_OFF);
            #pragma unroll
            for (int u = 0; u < 4; ++u) {
                int row = srow + 32 * u;
                async_copy_b128(x + (size_t)(blockRow + row) * DIM + knext + scg * 4,
                                xn + row * XS_STRIDE + scg * 4);
            }
            if (tid < RDIM * 8) {
                async_copy_b128(Wq + (size_t)srow * DIM + knext + scg * 4,
                                wn + srow * WS_STRIDE + scg * 4);
            }
        }

        #pragma unroll
        for (int kk = 0; kk < KTILE; kk += 4) {
            const int c0 = kk + 2 * hi;
            v2f a, bb;
            // A 16x4: lanes0-15 hold K=k..k+1, lanes16-31 hold K=k+2..k+3 for row M=ln
            a.x = xb[(mbase + ln) * XS_STRIDE + c0];
            a.y = xb[(mbase + ln) * XS_STRIDE + c0 + 1];
            // B 4x16: VGPR0 rows K={0,2}, VGPR1 rows K={1,3}; N = ln.  B[k][n] = Wq[n][k]
            bb.x = wb[ln * WS_STRIDE + c0];
            bb.y = wb[ln * WS_STRIDE + c0 + 1];
            acc = __builtin_amdgcn_wmma_f32_16x16x4_f32(
                false, a, false, bb, (short)0, acc, false, false);
        }

        wait_async0();      // this wave's prefetch for tile k0+32 has landed
        __syncthreads();    // everyone done reading cur buffer + all prefetches visible
    }

    // ---- scatter raw q (C layout: VGPR j -> row j + 8*hi, col ln) to LDS ----
    #pragma unroll
    for (int j = 0; j < 8; ++j) {
        int mloc = wid * 16 + j + 8 * hi;
        qs[mloc * QS_STRIDE + ln] = acc[j];
    }
    __syncthreads();
    // ---- per-row L2 normalize (F.normalize: v / max(||v||, eps)) ----
    if (tid < ROWS_PER_BLOCK) {
        float s = 0.f;
        #pragma unroll
        for (int r = 0; r < RDIM; ++r) { float v = qs[tid * QS_STRIDE + r]; s += v * v; }
        float inv = 1.0f / fmaxf(sqrtf(s), 1e-12f);
        #pragma unroll
        for (int r = 0; r < RDIM; ++r) qs[tid * QS_STRIDE + r] *= inv;
    }
    __syncthreads();

    // ---- scores = qn @ kn^T : 4 expert groups x 4 WMMA k-steps ----
    #pragma unroll
    for (int g = 0; g < 4; ++g) {
        v8f c2 = {};
        #pragma unroll
        for (int kk = 0; kk < RDIM; kk += 4) {
            const int c0 = kk + 2 * hi;
            v2f a, bb;
            a.x = qs[(wid * 16 + ln) * QS_STRIDE + c0];
            a.y = qs[(wid * 16 + ln) * QS_STRIDE + c0 + 1];
            bb.x = kn[(g * 16 + ln) * KN_STRIDE + c0];
            bb.y = kn[(g * 16 + ln) * KN_STRIDE + c0 + 1];
            c2 = __builtin_amdgcn_wmma_f32_16x16x4_f32(
                false, a, false, bb, (short)0, c2, false, false);
        }
        #pragma unroll
        for (int j = 0; j < 8; ++j) {
            int mloc = wid * 16 + j + 8 * hi;
            float v = c2[j];
            sc[mloc * SC_STRIDE + g * 16 + ln] = v;
            out_scores[(size_t)(blockRow + mloc) * NEXP + g * 16 + ln] = v;
        }
    }
    __syncthreads();

    // ---- top-2 + softmax, one row per thread ----
    if (tid < ROWS_PER_BLOCK) {
        const int m = blockRow + tid;
        float b1 = -__builtin_inff(), b2 = -__builtin_inff();
        int i1 = 0, i2 = 0;
        #pragma unroll 8
        for (int e = 0; e < NEXP; ++e) {
            float v = sc[tid * SC_STRIDE + e];
            if (v > b1)      { b2 = b1; i2 = i1; b1 = v; i1 = e; }
            else if (v > b2) { b2 = v; i2 = e; }
        }
        float d  = __expf(b2 - b1);          // stable 2-way softmax
        float w0 = 1.0f / (1.0f + d);
        out_idx[m * 2]     = i1;
        out_idx[m * 2 + 1] = i2;
        out_w[m * 2]       = w0;
        out_w[m * 2 + 1]   = d * w0;
    }
}

extern "C" void kernel_launch(void* const* d_in, const int* in_sizes, int n_in,
                              void* d_out, int out_size, void* d_ws, size_t ws_size,
                              hipStream_t stream) {
    const float* x    = (const float*)d_in[0];   // [16384, 2048]
    const float* Wq   = (const float*)d_in[1];   // [16, 2048]
    const float* keys = (const float*)d_in[2];   // [64, 16]

    // outputs concatenated flat in return order: topk_idx [N,2], weights [N,2], scores [N,64]
    int*   out_idx    = (int*)d_out;
    float* out_w      = (float*)d_out + (size_t)N_ROWS * 2;
    float* out_scores = (float*)d_out + (size_t)N_ROWS * 4;

    dim3 grid(N_ROWS / ROWS_PER_BLOCK);
    dim3 block(THREADS);
    cosine_router_kernel<<<grid, block, 0, stream>>>(x, Wq, keys, out_idx, out_w, out_scores);
}